// ThesisMarkowitzFullOpti_63436666962538
// MI455X (gfx1250) — compile-verified
//
#include <hip/hip_runtime.h>
#include <math.h>

#define N 128
#define NT 128                 // 4 wave32 per workgroup, one workgroup per sample
#define MAXW 0.2f
#define PGD_ITERS 200
#define BISECT_ITERS 40
#define EPSV 1e-8f

typedef __attribute__((ext_vector_type(2))) float v2f;
typedef __attribute__((ext_vector_type(8))) float v8f;

__device__ __forceinline__ float wave_sum(float x) {
    x += __shfl_xor(x, 1, 32);
    x += __shfl_xor(x, 2, 32);
    x += __shfl_xor(x, 4, 32);
    x += __shfl_xor(x, 8, 32);
    x += __shfl_xor(x, 16, 32);
    return x;
}
__device__ __forceinline__ float wave_min(float x) {
    x = fminf(x, __shfl_xor(x, 1, 32));
    x = fminf(x, __shfl_xor(x, 2, 32));
    x = fminf(x, __shfl_xor(x, 4, 32));
    x = fminf(x, __shfl_xor(x, 8, 32));
    x = fminf(x, __shfl_xor(x, 16, 32));
    return x;
}
__device__ __forceinline__ float wave_max(float x) {
    x = fmaxf(x, __shfl_xor(x, 1, 32));
    x = fmaxf(x, __shfl_xor(x, 2, 32));
    x = fmaxf(x, __shfl_xor(x, 4, 32));
    x = fmaxf(x, __shfl_xor(x, 8, 32));
    x = fmaxf(x, __shfl_xor(x, 16, 32));
    return x;
}
__device__ __forceinline__ float clipw(float x) {
    return fminf(fmaxf(x, -MAXW), MAXW);
}

__global__ __launch_bounds__(NT) void markowitz_qp_kernel(
    const float* __restrict__ rets,    // [B, N]
    const float* __restrict__ covmat,  // [B, N, N]
    const float* __restrict__ gamma,   // [B, 1]
    const float* __restrict__ alpha,   // [1]
    float* __restrict__ out)           // [B, N]
{
    extern __shared__ float smem[];
    float* Gs  = smem;             // N*N  : Gram matrix A = g^2 * M^T M (row-major)
    float* ws  = smem + N * N;     // N    : broadcast of w
    float* vs  = ws + N;           // N    : broadcast of v (pre-projection)
    float* red = vs + N;           // 8    : cross-wave partials

    const int b    = blockIdx.x;
    const int t    = threadIdx.x;      // 0..127, owns w[t]
    const int lane = t & 31;
    const int wave = t >> 5;
    const int lh   = lane >> 4;        // 0/1 : K-pair selector in WMMA operand layout
    const int l15  = lane & 15;        // M/N index within tile

    const float* M = covmat + (size_t)b * N * N;
    const float g  = gamma[b];
    const float g2 = g * g;                      // |g|^2 == g^2
    const float alpha_abs = fabsf(alpha[0]);

    // ---------------- Phase 1: Gs = g2 * (M^T M) via V_WMMA_F32_16X16X4_F32 ----------
    // Tile grid is 8x8; each wave computes 2 tile-rows (I = wave*32, wave*32+16).
    // Operand layout (ISA 7.12.2, 32-bit A 16x4 / B 4x16):
    //   element(lane, vgpr v) = M[kb + (lane>>4)*2 + v][tileBase + (lane&15)]
    for (int half = 0; half < 2; ++half) {
        const int I = wave * 32 + half * 16;
        v8f acc[8];
        #pragma unroll
        for (int j = 0; j < 8; ++j)
            #pragma unroll
            for (int p = 0; p < 8; ++p) acc[j][p] = 0.0f;

        for (int kb = 0; kb < N; kb += 4) {
            const int kr = kb + lh * 2;
            if (kb + 8 < N)   // pull upcoming rows from L2 (global_prefetch_b8)
                __builtin_prefetch(M + (size_t)(kr + 8) * N + l15, 0, 1);
            v2f a;
            a.x = M[(size_t)(kr + 0) * N + I + l15];
            a.y = M[(size_t)(kr + 1) * N + I + l15];
            #pragma unroll
            for (int j = 0; j < 8; ++j) {
                v2f bb;
                bb.x = M[(size_t)(kr + 0) * N + j * 16 + l15];
                bb.y = M[(size_t)(kr + 1) * N + j * 16 + l15];
                acc[j] = __builtin_amdgcn_wmma_f32_16x16x4_f32(
                    false, a, false, bb, (short)0, acc[j], false, false);
            }
        }
        // D layout: vgpr p -> (row = I + p + (lane>>4)*8, col = J*16 + (lane&15))
        #pragma unroll
        for (int j = 0; j < 8; ++j) {
            #pragma unroll
            for (int p = 0; p < 8; ++p) {
                const int row = I + p + lh * 8;
                const int col = j * 16 + l15;
                Gs[row * N + col] = acc[j][p] * g2;
            }
        }
    }
    __syncthreads();

    // ---------------- lr = 1 / (2*||A||_F + |alpha| + 1) -----------------------------
    float fr = 0.0f;
    #pragma unroll 4
    for (int k = 0; k < N; ++k) { const float gv = Gs[k * N + t]; fr += gv * gv; }
    fr = wave_sum(fr);
    if (lane == 0) red[wave] = fr;
    __syncthreads();
    const float fro2 = red[0] + red[1] + red[2] + red[3];
    const float lr = 1.0f / (2.0f * sqrtf(fro2) + alpha_abs + 1.0f);
    const float r_t = rets[(size_t)b * N + t];

    // ---------------- Phase 2: 200 projected-gradient iterations ---------------------
    float w_t = 1.0f / (float)N;
    for (int it = 0; it < PGD_ITERS; ++it) {
        ws[t] = w_t;
        __syncthreads();

        // ||w|| : each wave redundantly holds all 128 elems (4/lane) -> shuffle reduce
        const float wl0 = ws[lane], wl1 = ws[lane + 32],
                    wl2 = ws[lane + 64], wl3 = ws[lane + 96];
        const float wn = sqrtf(wave_sum(wl0 * wl0 + wl1 * wl1 +
                                        wl2 * wl2 + wl3 * wl3)) + EPSV;

        // Aw[t] = sum_k Gs[k][t] * w[k]  (column read of symmetric G: conflict-free)
        float aw = 0.0f;
        const float4* ws4 = (const float4*)ws;
        #pragma unroll 4
        for (int k4 = 0; k4 < N / 4; ++k4) {
            const float4 wv = ws4[k4];
            const int k = k4 * 4;
            aw += Gs[(k + 0) * N + t] * wv.x;
            aw += Gs[(k + 1) * N + t] * wv.y;
            aw += Gs[(k + 2) * N + t] * wv.z;
            aw += Gs[(k + 3) * N + t] * wv.w;
        }

        const float grad = r_t - 2.0f * aw - alpha_abs * w_t / wn;
        const float v_t = w_t + lr * grad;
        vs[t] = v_t;
        __syncthreads();

        // Simplex-box projection: per-wave redundant bisection, shuffle reductions only
        const float v0 = vs[lane], v1 = vs[lane + 32],
                    v2 = vs[lane + 64], v3 = vs[lane + 96];
        float lo = wave_min(fminf(fminf(v0, v1), fminf(v2, v3))) - MAXW;
        float hi = wave_max(fmaxf(fmaxf(v0, v1), fmaxf(v2, v3))) + MAXW;
        for (int bi = 0; bi < BISECT_ITERS; ++bi) {
            const float mid = 0.5f * (lo + hi);
            const float s = wave_sum(clipw(v0 - mid) + clipw(v1 - mid) +
                                     clipw(v2 - mid) + clipw(v3 - mid));
            if (s > 1.0f) lo = mid; else hi = mid;
        }
        const float nu = 0.5f * (lo + hi);
        w_t = clipw(v_t - nu);
        // no extra barrier needed: next ws[t] write is ordered behind the vs barrier
    }

    out[(size_t)b * N + t] = w_t;
}

extern "C" void kernel_launch(void* const* d_in, const int* in_sizes, int n_in,
                              void* d_out, int out_size, void* d_ws, size_t ws_size,
                              hipStream_t stream) {
    const float* rets   = (const float*)d_in[0];
    const float* covmat = (const float*)d_in[1];
    const float* gamma  = (const float*)d_in[2];
    const float* alpha  = (const float*)d_in[3];
    float* out = (float*)d_out;

    const int B = in_sizes[0] / N;   // 2048
    const size_t shmem = (size_t)(N * N + N + N + 8) * sizeof(float);  // ~65.5 KB

    // Allow >64KB dynamic LDS (gfx1250 WGP has 320KB); ignore result if unsupported.
    (void)hipFuncSetAttribute((const void*)markowitz_qp_kernel,
                              hipFuncAttributeMaxDynamicSharedMemorySize,
                              (int)shmem);

    markowitz_qp_kernel<<<B, NT, shmem, stream>>>(rets, covmat, gamma, alpha, out);
}